// rdf_56521769615647
// MI455X (gfx1250) — compile-verified
//
#include <hip/hip_runtime.h>

typedef float v2f __attribute__((ext_vector_type(2)));
typedef float v8f __attribute__((ext_vector_type(8)));

constexpr int kN      = 2048;
constexpr int kBins   = 100;
constexpr int kTile   = 16;
constexpr int kITiles = kN / kTile;   // 128 tiles of 16 atoms
constexpr int kJSplit = 4;            // grid.y
constexpr int kWaves  = 8;            // 256 threads / 32
constexpr int kTilesPerWave = kITiles / (kJSplit * kWaves); // 4

__global__ __launch_bounds__(256)
void rdf_pair_kernel(const float* __restrict__ xyz,
                     const float* __restrict__ cell,
                     const float* __restrict__ offsets,
                     float* __restrict__ ws_count)
{
    __shared__ float4 s_pos[kN];     // x,y,z,|r|^2  (32 KB)
    __shared__ float  s_cnt[kBins];

    const int  tid  = threadIdx.x;
    const int  lane = tid & 31;
    const int  wave = tid >> 5;
    const bool lo   = lane < 16;
    const int  ln16 = lane & 15;

    // Stage all coordinates (+ squared norms) into LDS once per WG.
    for (int a = tid; a < kN; a += 256) {
        float x = xyz[3 * a + 0];
        float y = xyz[3 * a + 1];
        float z = xyz[3 * a + 2];
        s_pos[a] = make_float4(x, y, z, fmaf(x, x, fmaf(y, y, z * z)));
    }
    if (tid < kBins) s_cnt[tid] = 0.0f;
    __syncthreads();

    const float L  = cell[0];
    const float width = offsets[1] - offsets[0];
    // exp(coeff*x) == exp2(c2*x), coeff = -0.5/width^2
    const float c2 = (-0.5f / (width * width)) * 1.44269504088896340736f;

    // Lane 'l' accumulates bins {l, l+32, l+64, l+96}.
    const float o0 = width * (float)lane;
    const float o1 = o0 + 32.0f * width;
    const float o2 = o0 + 64.0f * width;
    const float o3 = (lane < 4) ? (o0 + 96.0f * width) : 1.0e9f; // 1e9 -> exp2 underflows to 0
    float acc0 = 0.0f, acc1 = 0.0f, acc2 = 0.0f, acc3 = 0.0f;

    // A fragment (i-tile fixed per workgroup): 16x4 f32, K = {x,y,z,0}
    // lanes 0-15 hold K=0,1 ; lanes 16-31 hold K=2,3 (ISA 7.12.2)
    const int ibase = blockIdx.x * kTile;
    const float4 pa = s_pos[ibase + ln16];
    v2f afrag;
    afrag.x = lo ? pa.x : pa.z;
    afrag.y = lo ? pa.y : 0.0f;

    // C layout: VGPR p -> M = p (lanes 0-15) / p+8 (lanes 16-31), N = lane&15.
    // The 8 i-rows this lane will pair against are tile-invariant: hoist them.
    float4 pi[8];
#pragma unroll
    for (int p = 0; p < 8; ++p) pi[p] = s_pos[ibase + p + (lo ? 0 : 8)];

    for (int t = 0; t < kTilesPerWave; ++t) {
        const int jt    = (blockIdx.y * kWaves + wave) * kTilesPerWave + t;
        const int jbase = jt * kTile;
        const int jglob = jbase + ln16;

        // B fragment: 4x16 f32; lanes 0-15 hold K rows 0,1 ; lanes 16-31 rows 2,3
        const float4 pj = s_pos[jglob];
        v2f bfrag;
        bfrag.x = lo ? pj.x : pj.z;
        bfrag.y = lo ? pj.y : 0.0f;

        // S[m][n] = xi[m] . xj[n] for the 16x16 pair tile
        v8f cm = {};
        cm = __builtin_amdgcn_wmma_f32_16x16x4_f32(false, afrag, false, bfrag,
                                                   (short)0, cm, false, false);

#pragma unroll
        for (int p = 0; p < 8; ++p) {
            const int iglob = ibase + p + (lo ? 0 : 8);

            // exact branch-free minimum image: |dxw| = min(|dx|, L - |dx|)
            const float ax  = __builtin_fabsf(pi[p].x - pj.x);
            const float ay  = __builtin_fabsf(pi[p].y - pj.y);
            const float az  = __builtin_fabsf(pi[p].z - pj.z);
            const float axw = fminf(ax, L - ax);
            const float ayw = fminf(ay, L - ay);
            const float azw = fminf(az, L - az);
            const float dsq = fmaf(axw, axw, fmaf(ayw, ayw, azw * azw));

            // WMMA-based unwrapped dsq used as a (always-consistent) prefilter:
            // in-cutoff => each |dx| < 8 + L = 28 => dsq_g < 3*28^2 = 2352 < 2400.
            const float dsq_g = fmaf(-2.0f, cm[p], pi[p].w + pj.w);

            const bool act = (iglob != jglob) && (dsq > 0.0f) && (dsq < 64.0f) &&
                             (dsq_g < 2400.0f);
            // dsq in (0,64] for active pairs: raw v_sqrt_f32 (~1 ulp) is plenty
            const float d = __builtin_amdgcn_sqrtf(dsq);

            // broadcast each in-cutoff distance; every lane covers its 4 bins
            unsigned mb = (unsigned)__ballot(act);
            while (mb) {
                const int src = __builtin_ctz(mb);
                mb &= mb - 1u;
                const float ds =
                    __uint_as_float(__builtin_amdgcn_readlane(__float_as_uint(d), src));
                float f0 = ds - o0, f1 = ds - o1, f2 = ds - o2, f3 = ds - o3;
                acc0 += __builtin_amdgcn_exp2f(c2 * f0 * f0);
                acc1 += __builtin_amdgcn_exp2f(c2 * f1 * f1);
                acc2 += __builtin_amdgcn_exp2f(c2 * f2 * f2);
                acc3 += __builtin_amdgcn_exp2f(c2 * f3 * f3);
            }
        }
    }

    atomicAdd(&s_cnt[lane],      acc0);
    atomicAdd(&s_cnt[lane + 32], acc1);
    atomicAdd(&s_cnt[lane + 64], acc2);
    if (lane < 4) atomicAdd(&s_cnt[lane + 96], acc3);
    __syncthreads();
    if (tid < kBins) unsafeAtomicAdd(&ws_count[tid], s_cnt[tid]);
}

__global__ __launch_bounds__(128)
void rdf_norm_kernel(const float* __restrict__ ws_count,
                     const float* __restrict__ bins,
                     float* __restrict__ out)
{
    __shared__ float red[128];
    const int t = threadIdx.x;
    const float v = (t < kBins) ? ws_count[t] : 0.0f;
    red[t] = v;
    __syncthreads();
    for (int s = 64; s > 0; s >>= 1) {
        if (t < s) red[t] += red[t + s];
        __syncthreads();
    }
    const float total = red[0];

    if (t <= kBins) out[kBins + t] = bins[t];          // bins copy: out[100..200]
    if (t < kBins) {
        const float cn = v / total;
        out[t] = cn;                                    // normalized count: out[0..99]
        const float pi43 = (4.0f / 3.0f) * 3.14159265358979323846f;
        const float b0 = bins[t], b1 = bins[t + 1];
        const float vol = pi43 * (b1 * b1 * b1 - b0 * b0 * b0);
        const float V   = pi43 * (7.5f * 7.5f * 7.5f);
        out[2 * kBins + 1 + t] = cn * (V / vol);        // rdf: out[201..300]
    }
}

extern "C" void kernel_launch(void* const* d_in, const int* in_sizes, int n_in,
                              void* d_out, int out_size, void* d_ws, size_t ws_size,
                              hipStream_t stream)
{
    (void)in_sizes; (void)n_in; (void)out_size; (void)ws_size;
    const float* xyz     = (const float*)d_in[0];
    const float* cell    = (const float*)d_in[1];
    const float* bins    = (const float*)d_in[2];
    const float* offsets = (const float*)d_in[3];
    float* out = (float*)d_out;
    float* ws  = (float*)d_ws;

    (void)hipMemsetAsync(ws, 0, kBins * sizeof(float), stream);
    rdf_pair_kernel<<<dim3(kITiles, kJSplit), 256, 0, stream>>>(xyz, cell, offsets, ws);
    rdf_norm_kernel<<<1, 128, 0, stream>>>(ws, bins, out);
}